// MeshLoss2D_15857019256897
// MI455X (gfx1250) — compile-verified
//
#include <hip/hip_runtime.h>
#include <hip/hip_bf16.h>

typedef __attribute__((ext_vector_type(2))) float v2f;
typedef __attribute__((ext_vector_type(8))) float v8f;

#define BB   4          // batch
#define ZZ   3          // coords
#define GX   28         // input mesh x
#define GY   28         // input mesh y
#define NXR  82         // refined x = (28-1)*3+1
#define NYR  82         // refined y
#define MM   (NXR*NYR)  // 6724 mesh points
#define MP   6752       // padded to 422*16 (even tile count for 2x unroll)
#define MT   (MP/16)    // 422 m-tiles
#define MT2  (MT/2)     // 211 tile-pairs
#define NN   8192       // pc points per batch
#define NWAVES (BB*(NN/32))  // 1024 work items (wave handles 32 pc pts)

// ---------------------------------------------------------------------------
// Prep 1: refine mesh (bilinear, align-corners, factor 3) and emit A-matrix
// rows: meshA[b][m] = (x, y, z, x^2+y^2+z^2). Padding rows get m2=1e30 so
// they can never win the min-reduction.
// ---------------------------------------------------------------------------
__global__ void prep_mesh_kernel(const float* __restrict__ mesh,
                                 float* __restrict__ meshA) {
    int idx = blockIdx.x * blockDim.x + threadIdx.x;
    if (idx >= BB * MP) return;
    int b = idx / MP;
    int m = idx - b * MP;
    float o0 = 0.f, o1 = 0.f, o2 = 0.f, o3 = 1.0e30f;
    if (m < MM) {
        int xi = m / NYR, yi = m - (m / NYR) * NYR;
        int jx = min(xi / 3, GX - 2);
        int jy = min(yi / 3, GY - 2);
        float wx = (float)(xi - 3 * jx) * (1.0f / 3.0f);
        float wy = (float)(yi - 3 * jy) * (1.0f / 3.0f);
        float c[ZZ];
#pragma unroll
        for (int z = 0; z < ZZ; ++z) {
            const float* p = mesh + (((size_t)b * ZZ + z) * GX + jx) * GY + jy;
            float v00 = p[0], v01 = p[1], v10 = p[GY], v11 = p[GY + 1];
            float top = v00 + wy * (v01 - v00);   // interp along y
            float bot = v10 + wy * (v11 - v10);
            c[z] = top + wx * (bot - top);        // interp along x
        }
        o0 = c[0]; o1 = c[1]; o2 = c[2];
        o3 = c[0] * c[0] + c[1] * c[1] + c[2] * c[2];
    }
    float* o = meshA + (size_t)idx * 4;
    o[0] = o0; o[1] = o1; o[2] = o2; o[3] = o3;
}

// ---------------------------------------------------------------------------
// Prep 2: B-matrix columns pcB[b][n] = (-2x, -2y, -2z, 1) and n2[b][n] = |p|^2
// ---------------------------------------------------------------------------
__global__ void prep_pc_kernel(const float* __restrict__ pc,
                               float* __restrict__ pcB,
                               float* __restrict__ n2) {
    int idx = blockIdx.x * blockDim.x + threadIdx.x;
    if (idx >= BB * NN) return;
    int b = idx / NN;
    int n = idx - b * NN;
    float x = pc[((size_t)b * ZZ + 0) * NN + n];
    float y = pc[((size_t)b * ZZ + 1) * NN + n];
    float z = pc[((size_t)b * ZZ + 2) * NN + n];
    float* o = pcB + (size_t)idx * 4;
    o[0] = -2.0f * x; o[1] = -2.0f * y; o[2] = -2.0f * z; o[3] = 1.0f;
    n2[idx] = x * x + y * y + z * z;
}

// Raw v_min3_num_f32: no NaN-canonicalize legalization (data is NaN-free),
// folds running-min + two fresh WMMA results in a single VALU op.
__device__ __forceinline__ float min3_raw(float a, float b, float c) {
    float r;
    asm("v_min3_num_f32 %0, %1, %2, %3" : "=v"(r) : "v"(a), "v"(b), "v"(c));
    return r;
}

// ---------------------------------------------------------------------------
// Main: one wave per 32 pc points. Loop over 211 M-tile pairs; per iteration:
//   2 pipelined A-fragment loads (b64), 4x v_wmma_f32_16x16x4_f32 (C=0)
//   producing D[m,n] = m2 - 2*cross, and 16 v_min3_num_f32 that fold both
//   tiles of a pair into the running elementwise min (4 VALU per WMMA).
// A-fragment layout (ISA 7.12.2, f32 16x4): lane<16 -> row M=lane, K0/K1;
// lane>=16 -> row M=lane-16, K2/K3  => one b64 load from meshA[m][half*2].
// B 4x16 mirrors this from pcB[n][half*2]. C/D: lane<16 holds rows 0..7 of
// column N=lane; lane+16 holds rows 8..15 -> shfl_xor(16) to combine.
// ---------------------------------------------------------------------------
__global__ __launch_bounds__(256) void chamfer_wmma_kernel(
        const float* __restrict__ meshA, const float* __restrict__ pcB,
        const float* __restrict__ n2, float* __restrict__ partial) {
    const int lane = threadIdx.x & 31;
    const int wid  = blockIdx.x * (blockDim.x >> 5) + (threadIdx.x >> 5);
    const int b    = wid >> 8;          // 256 work items per batch
    const int nt   = wid & 255;
    const int n_base = nt * 32;
    const int half = lane >> 4;
    const int l16  = lane & 15;

    // Two B fragments (columns n_base..n_base+15, n_base+16..n_base+31)
    const float* pcb = pcB + ((size_t)b * NN + n_base) * 4;
    const v2f bf0 = *(const v2f*)(pcb + (size_t)l16 * 4        + half * 2);
    const v2f bf1 = *(const v2f*)(pcb + (size_t)(16 + l16) * 4 + half * 2);

    const float* Ab = meshA + (size_t)b * MP * 4 + (size_t)l16 * 4 + half * 2;
    v8f run0, run1;
#pragma unroll
    for (int i = 0; i < 8; ++i) { run0[i] = 3.0e38f; run1[i] = 3.0e38f; }

    v2f af0 = *(const v2f*)(Ab);              // tile 0
    v2f af1 = *(const v2f*)(Ab + 64);         // tile 1
    for (int mt2 = 0; mt2 < MT2; ++mt2) {
        // Issue next pair's fragment loads before consuming current pair.
        v2f n0 = af0, n1 = af1;
        if (mt2 + 1 < MT2) {
            const float* nxt = Ab + (size_t)(2 * mt2 + 2) * 64;
            n0 = *(const v2f*)(nxt);
            n1 = *(const v2f*)(nxt + 64);
        }
        if (mt2 + 4 < MT2)                    // keep A stream warm near WGP
            __builtin_prefetch(Ab + (size_t)(2 * mt2 + 8) * 64, 0, 3);

        v8f z8 = {};
        v8f dA0 = __builtin_amdgcn_wmma_f32_16x16x4_f32(
            false, af0, false, bf0, (short)0, z8, false, false);
        v8f dB0 = __builtin_amdgcn_wmma_f32_16x16x4_f32(
            false, af1, false, bf0, (short)0, z8, false, false);
        v8f dA1 = __builtin_amdgcn_wmma_f32_16x16x4_f32(
            false, af0, false, bf1, (short)0, z8, false, false);
        v8f dB1 = __builtin_amdgcn_wmma_f32_16x16x4_f32(
            false, af1, false, bf1, (short)0, z8, false, false);
#pragma unroll
        for (int i = 0; i < 8; ++i) run0[i] = min3_raw(run0[i], dA0[i], dB0[i]);
#pragma unroll
        for (int i = 0; i < 8; ++i) run1[i] = min3_raw(run1[i], dA1[i], dB1[i]);

        af0 = n0;
        af1 = n1;
    }

    // 8 -> 1 per-lane reduction (cold path).
    float r0 = fminf(fminf(fminf(run0[0], run0[1]), fminf(run0[2], run0[3])),
                     fminf(fminf(run0[4], run0[5]), fminf(run0[6], run0[7])));
    float r1 = fminf(fminf(fminf(run1[0], run1[1]), fminf(run1[2], run1[3])),
                     fminf(fminf(run1[4], run1[5]), fminf(run1[6], run1[7])));

    // Combine rows 0..7 (lane) with rows 8..15 (lane^16) for each column.
    r0 = fminf(r0, __shfl_xor(r0, 16, 32));
    r1 = fminf(r1, __shfl_xor(r1, 16, 32));

    float contrib = 0.0f;
    if (lane < 16) {
        const float* n2b = n2 + (size_t)b * NN + n_base;
        contrib = (r0 + n2b[l16]) + (r1 + n2b[16 + l16]);
    }
#pragma unroll
    for (int off = 16; off > 0; off >>= 1)
        contrib += __shfl_xor(contrib, off, 32);
    if (lane == 0) partial[wid] = contrib;
}

// ---------------------------------------------------------------------------
// Deterministic final reduction: 1 block sums 1024 partials, scales by 1/(B*N)
// ---------------------------------------------------------------------------
__global__ __launch_bounds__(256) void final_reduce_kernel(
        const float* __restrict__ partial, float* __restrict__ out) {
    __shared__ float s[256];
    float sum = 0.0f;
    for (int i = threadIdx.x; i < NWAVES; i += 256) sum += partial[i];
    s[threadIdx.x] = sum;
    __syncthreads();
    for (int o = 128; o > 0; o >>= 1) {
        if (threadIdx.x < o) s[threadIdx.x] += s[threadIdx.x + o];
        __syncthreads();
    }
    if (threadIdx.x == 0) out[0] = s[0] / (float)(BB * NN);
}

extern "C" void kernel_launch(void* const* d_in, const int* in_sizes, int n_in,
                              void* d_out, int out_size, void* d_ws, size_t ws_size,
                              hipStream_t stream) {
    const float* mesh = (const float*)d_in[0];  // [4,3,28,28]
    const float* pc   = (const float*)d_in[1];  // [4,3,8192]
    float* out = (float*)d_out;

    // Workspace carve-up (256B aligned)
    char* ws = (char*)d_ws;
    size_t off = 0;
    float* meshA = (float*)(ws + off); off += (size_t)BB * MP * 4 * sizeof(float);
    off = (off + 255) & ~(size_t)255;
    float* pcB   = (float*)(ws + off); off += (size_t)BB * NN * 4 * sizeof(float);
    off = (off + 255) & ~(size_t)255;
    float* n2    = (float*)(ws + off); off += (size_t)BB * NN * sizeof(float);
    off = (off + 255) & ~(size_t)255;
    float* partial = (float*)(ws + off); off += (size_t)NWAVES * sizeof(float);
    (void)ws_size; (void)in_sizes; (void)n_in; (void)out_size;

    prep_mesh_kernel<<<(BB * MP + 255) / 256, 256, 0, stream>>>(mesh, meshA);
    prep_pc_kernel<<<(BB * NN + 255) / 256, 256, 0, stream>>>(pc, pcB, n2);
    chamfer_wmma_kernel<<<NWAVES / 8, 256, 0, stream>>>(meshA, pcB, n2, partial);
    final_reduce_kernel<<<1, 256, 0, stream>>>(partial, out);
}